// Conv2D_17454747091359
// MI455X (gfx1250) — compile-verified
//
#include <hip/hip_runtime.h>

// CDNA5 / gfx1250: wave32, V_WMMA_F32_16X16X4_F32 (D = A(16x4) * B(4x16) + C)
typedef __attribute__((ext_vector_type(2))) float v2f;
typedef __attribute__((ext_vector_type(8))) float v8f;

#define IH 6144
#define IW 6144
#define OH 6138
#define OW 6138
#define KH 7
#define KW 7
#define NCHUNK 6            // K per ky padded to 24 -> 6 chunks of 4
#define TILES_PER_WAVE 4    // vertically stacked 16x16 tiles per wave

// out(y0+i, x0+j) = sum_ky sum_u in(y0+i+ky, x0+u) * Wt_ky(u, j)
// Wt_ky(u,j) = w(ky, u-j) for 0 <= u-j <= 6, else 0  (banded Toeplitz).
// B fragments are tile-invariant -> precomputed once per block into LDS.
__global__ __launch_bounds__(256) void conv7x7_wmma_f32(
    const float* __restrict__ x,
    const float* __restrict__ w,
    const float* __restrict__ bias,
    float* __restrict__ out) {
  __shared__ float sw[KH * KW];
  __shared__ v2f sB[KH * NCHUNK * 32];  // [chunk=(ky*6+uc)][lane] -> (b.x, b.y)

  const int tid = threadIdx.x;
  if (tid < KH * KW) sw[tid] = w[tid];
  __syncthreads();

  // Branchless B-table build: clamped LDS read * {0,1} mask (v_cndmask, no exec ops).
  for (int t = tid; t < KH * NCHUNK * 32; t += 256) {
    const int ln = t & 31;
    const int chunk = t >> 5;          // 0..41
    const int uc = chunk % NCHUNK;
    const int ky = chunk / NCHUNK;
    const int j = ln & 15;             // N column
    const int khf = ln >> 4;           // K-half selector
    const int d0 = uc * 4 + khf * 2 - j;
    const int c0 = d0 < 0 ? 0 : (d0 > 6 ? 6 : d0);
    const int c1 = d0 + 1 < 0 ? 0 : (d0 + 1 > 6 ? 6 : d0 + 1);
    const float m0 = (d0 >= 0 && d0 < KW) ? 1.0f : 0.0f;
    const float m1 = (d0 + 1 >= 0 && d0 + 1 < KW) ? 1.0f : 0.0f;
    v2f b;
    b.x = sw[ky * KW + c0] * m0;
    b.y = sw[ky * KW + c1] * m1;
    sB[t] = b;
  }
  __syncthreads();

  const int lane = tid & 31;
  const int wave = tid >> 5;
  const int i = lane & 15;   // A: M row in tile; B/C/D: N column
  const int kh = lane >> 4;  // K-half selector
  const int tx = blockIdx.x * 8 + wave;  // tile col 0..383
  const int x0 = tx << 4;
  const float b0 = bias[0];
  const v2f* bt = &sB[lane];  // chunk stride = 32

#pragma unroll 1
  for (int t = 0; t < TILES_PER_WAVE; ++t) {
    const int ty = blockIdx.y * TILES_PER_WAVE + t;  // tile row 0..383
    const int y0 = ty << 4;
    v8f acc = {};

    // Wave-uniform interior test (EXEC stays all-ones through WMMAs).
    if (y0 + 21 < IH && x0 + 23 < IW) {
      // One per-lane base; all 42 loads are base + compile-time immediate.
      const float* bp = x + (long)(y0 + i) * IW + x0 + kh * 2;
#pragma unroll
      for (int ky = 0; ky < KH; ++ky) {
#pragma unroll
        for (int uc = 0; uc < NCHUNK; ++uc) {
          const v2f a = *reinterpret_cast<const v2f*>(bp + ky * IW + uc * 4);
          const v2f b = bt[(ky * NCHUNK + uc) * 32];
          acc = __builtin_amdgcn_wmma_f32_16x16x4_f32(
              false, a, false, b, (short)0, acc, false, false);
        }
      }
      // Interior tiles: rows <= 6127, cols <= 6127 -> unguarded stores.
#pragma unroll
      for (int v = 0; v < 8; ++v)
        out[(long)(y0 + v + kh * 8) * OW + x0 + i] = acc[v] + b0;
    } else {
      // Edge tiles (last tile row/col): clamped addresses; clamped garbage
      // only feeds output rows/cols >= 6138, which are never stored.
#pragma unroll
      for (int ky = 0; ky < KH; ++ky) {
#pragma unroll
        for (int uc = 0; uc < NCHUNK; ++uc) {
          int row = y0 + i + ky;
          row = row > IH - 1 ? IH - 1 : row;
          int col = x0 + uc * 4 + kh * 2;
          col = col > IW - 2 ? IW - 2 : col;
          const v2f a = *reinterpret_cast<const v2f*>(x + (long)row * IW + col);
          const v2f b = bt[(ky * NCHUNK + uc) * 32];
          acc = __builtin_amdgcn_wmma_f32_16x16x4_f32(
              false, a, false, b, (short)0, acc, false, false);
        }
      }
#pragma unroll
      for (int v = 0; v < 8; ++v) {
        const int orow = y0 + v + kh * 8;
        const int ocol = x0 + i;
        if (orow < OH && ocol < OW)
          out[(long)orow * OW + ocol] = acc[v] + b0;
      }
    }
  }
}

extern "C" void kernel_launch(void* const* d_in, const int* in_sizes, int n_in,
                              void* d_out, int out_size, void* d_ws, size_t ws_size,
                              hipStream_t stream) {
  const float* x    = (const float*)d_in[0];
  const float* w    = (const float*)d_in[1];
  const float* bias = (const float*)d_in[2];
  float* out        = (float*)d_out;

  // 384x384 tiles of 16x16; 8 waves (tile columns) per block, 4 tiles per wave.
  dim3 grid(384 / 8, 384 / TILES_PER_WAVE);
  conv7x7_wmma_f32<<<grid, 256, 0, stream>>>(x, w, bias, out);
}